// KLDLoss_17987323036509
// MI455X (gfx1250) — compile-verified
//
#include <hip/hip_runtime.h>

typedef float v4f __attribute__((ext_vector_type(4)));
typedef float v2f __attribute__((ext_vector_type(2)));
typedef float v8f __attribute__((ext_vector_type(8)));

#define WAVE   32
#define BLOCK  256
#define MAXBLK 4096

// Full-wave (32-lane) sum using V_WMMA_F32_16X16X4_F32 with B = ones.
// A layout (32-bit 16x4 MxK): VGPR0 = {lanes 0-15: K=0, lanes 16-31: K=2},
// VGPR1 = {K=1, K=3}. Putting acc in VGPR0 and 0 in VGPR1 gives
// rowsum_m = acc_m + acc_{m+16}. D layout: VGPR r holds M=r (lanes 0-15)
// and M=r+8 (lanes 16-31), so sum(c[0..7]) gives half-sums; one
// shfl_xor(16) completes the 32-lane total in every lane.
// EXEC is all 1s at the call site (uniform post-loop control flow).
__device__ __forceinline__ float wave_sum_wmma(float acc) {
    v2f a; a[0] = acc;  a[1] = 0.0f;
    v2f b; b[0] = 1.0f; b[1] = 1.0f;
    v8f c = {0.f, 0.f, 0.f, 0.f, 0.f, 0.f, 0.f, 0.f};
    c = __builtin_amdgcn_wmma_f32_16x16x4_f32(
            /*neg_a=*/false, a, /*neg_b=*/false, b,
            /*c_mod=*/(short)0, c, /*reuse_a=*/false, /*reuse_b=*/false);
    float s = ((c[0] + c[1]) + (c[2] + c[3])) + ((c[4] + c[5]) + (c[6] + c[7]));
    s += __shfl_xor(s, 16, WAVE);
    return s;
}

__device__ __forceinline__ float kl_elem(float mq, float sq, float mp, float sp) {
    float d  = mq - mp;
    float ds = sp - sq;
    // (sq - sp) + exp(sp - sq) + d*d*exp(-sq)
    return (-ds) + __expf(ds) + d * d * __expf(-sq);
}

__global__ void kl_partial_kernel(const float* __restrict__ mu_q,
                                  const float* __restrict__ sg_q,
                                  const float* __restrict__ mu_p,
                                  const float* __restrict__ sg_p,
                                  float* __restrict__ partials,
                                  long long n) {
    const long long n4     = n >> 2;
    const long long tid    = (long long)blockIdx.x * blockDim.x + threadIdx.x;
    const long long stride = (long long)gridDim.x * blockDim.x;

    const v4f* __restrict__ mq4 = (const v4f*)mu_q;
    const v4f* __restrict__ sq4 = (const v4f*)sg_q;
    const v4f* __restrict__ mp4 = (const v4f*)mu_p;
    const v4f* __restrict__ sp4 = (const v4f*)sg_p;

    float acc = 0.0f;

    // Main stream: 4 x global_load_b128 (non-temporal: single-pass > L2) per iter.
    for (long long i = tid; i < n4; i += stride) {
        v4f mq = __builtin_nontemporal_load(&mq4[i]);
        v4f sq = __builtin_nontemporal_load(&sq4[i]);
        v4f mp = __builtin_nontemporal_load(&mp4[i]);
        v4f sp = __builtin_nontemporal_load(&sp4[i]);
#pragma unroll
        for (int k = 0; k < 4; ++k)
            acc += kl_elem(mq[k], sq[k], mp[k], sp[k]);
    }
    // Scalar tail (n % 4), normally empty for D=2048.
    for (long long i = (n4 << 2) + tid; i < n; i += stride)
        acc += kl_elem(mu_q[i], sg_q[i], mu_p[i], sg_p[i]);

    // Wave reduction via WMMA (uniform control flow here -> EXEC all 1s).
    float wsum = wave_sum_wmma(acc);

    __shared__ float smem[BLOCK / WAVE];
    const int lane = threadIdx.x & (WAVE - 1);
    const int wid  = threadIdx.x >> 5;
    if (lane == 0) smem[wid] = wsum;
    __syncthreads();
    if (threadIdx.x == 0) {
        float s = 0.0f;
#pragma unroll
        for (int w = 0; w < BLOCK / WAVE; ++w) s += smem[w];
        partials[blockIdx.x] = s;
    }
}

__global__ void kl_finalize_kernel(const float* __restrict__ partials, int nparts,
                                   float* __restrict__ out, double totalN) {
    double acc = 0.0;
    for (int i = (int)threadIdx.x; i < nparts; i += blockDim.x)
        acc += (double)partials[i];

#pragma unroll
    for (int off = 16; off > 0; off >>= 1)
        acc += __shfl_xor(acc, off, WAVE);

    __shared__ double smem[BLOCK / WAVE];
    const int lane = threadIdx.x & (WAVE - 1);
    const int wid  = threadIdx.x >> 5;
    if (lane == 0) smem[wid] = acc;
    __syncthreads();
    if (threadIdx.x == 0) {
        double t = 0.0;
#pragma unroll
        for (int w = 0; w < BLOCK / WAVE; ++w) t += smem[w];
        out[0] = (float)(0.5 * (t - totalN));
    }
}

extern "C" void kernel_launch(void* const* d_in, const int* in_sizes, int n_in,
                              void* d_out, int out_size, void* d_ws, size_t ws_size,
                              hipStream_t stream) {
    const float* mu_q = (const float*)d_in[0];
    const float* sg_q = (const float*)d_in[1];
    const float* mu_p = (const float*)d_in[2];
    const float* sg_p = (const float*)d_in[3];
    const long long n = (long long)in_sizes[0];  // B*D

    float* partials = (float*)d_ws;

    int nblocks = MAXBLK;
    const long long maxByWs = (long long)(ws_size / sizeof(float));
    if ((long long)nblocks > maxByWs) nblocks = (int)maxByWs;
    const long long needed = (n / 4 + BLOCK - 1) / BLOCK;
    if ((long long)nblocks > needed) nblocks = (int)needed;
    if (nblocks < 1) nblocks = 1;

    kl_partial_kernel<<<nblocks, BLOCK, 0, stream>>>(mu_q, sg_q, mu_p, sg_p,
                                                     partials, n);
    kl_finalize_kernel<<<1, BLOCK, 0, stream>>>(partials, nblocks,
                                                (float*)d_out, (double)n);
}